// GraphTrans_41669772706028
// MI455X (gfx1250) — compile-verified
//
#include <hip/hip_runtime.h>
#include <hip/hip_bf16.h>
#include <math.h>

// ---------------- problem constants ----------------
#define NN    10000      // nodes
#define EE    160000     // edges (before self-loops)
#define ET    170000     // edges + self-loops
#define DIN   384
#define F1    512        // H1*C1
#define H1    2
#define C1    256
#define D2    384        // layer-2 feature dim (heads=1)

#define KC    32         // K-chunk staged in LDS per TDM transfer
#define NFMAX 512        // max Nf staged (GEMM1)

typedef float v2f __attribute__((ext_vector_type(2)));
typedef float v8f __attribute__((ext_vector_type(8)));
typedef unsigned int v4u __attribute__((ext_vector_type(4)));
typedef int v4i __attribute__((ext_vector_type(4)));
typedef int v8i __attribute__((ext_vector_type(8)));

// ---------------- helpers ----------------
__device__ __forceinline__ void atomicMaxF(float* addr, float val) {
    // Monotonic IEEE-754 bit trick: signed max for >=0, unsigned min for <0.
    if (val >= 0.0f) atomicMax((int*)addr, __float_as_int(val));
    else             atomicMin((unsigned int*)addr, __float_as_uint(val));
}

__device__ __forceinline__ unsigned lds_off(const void* p) {
    // LDS aperture: flat address bits [31:0] are the LDS byte offset.
    return (unsigned)(uintptr_t)p;
}

// Issue one TDM 2-D tile load: rows x row_len f32 elements, row stride
// row_stride elements, global -> LDS.  D# per CDNA5 ISA
// (group0: count=1,type=2; group1: data_size=4B, 2-D tile; groups 2/3 and
// the extra group zero).  6-arg builtin form (amdgpu-toolchain clang-23).
__device__ __forceinline__ void tdm_load_2d(const void* gaddr, unsigned ldsoff,
                                            unsigned rows, unsigned row_len,
                                            unsigned long long row_stride) {
    unsigned long long ga = (unsigned long long)(uintptr_t)gaddr;
    v4u g0;
    g0.x = 1u;                                   // count=1, user descriptor
    g0.y = ldsoff;                               // lds_addr[31:0]
    g0.z = (unsigned)ga;                         // global_addr[31:0]
    g0.w = (unsigned)(ga >> 32) | (2u << 30);    // global_addr[56:32] | type=2
    unsigned td0 = row_len;                      // tensor_dim0 (elements/row)
    unsigned td1 = rows;                         // tensor_dim1 (rows)
    v8i g1;
    g1[0] = (int)(2u << 16);                     // wg_mask=0, data_size=2 (4B)
    g1[1] = (int)((td0 & 0xFFFFu) << 16);        // tensor_dim0[15:0]
    g1[2] = (int)((td0 >> 16) | ((td1 & 0xFFFFu) << 16));
    g1[3] = (int)((td1 >> 16) | (row_len << 16));// tile_dim0 = row_len
    g1[4] = (int)rows;                           // tile_dim1 = rows, tile_dim2=0
    g1[5] = (int)(unsigned)(row_stride & 0xFFFFFFFFull);   // dim0_stride[31:0]
    g1[6] = (int)(unsigned)(row_stride >> 32);             // dim0_stride[47:32]
    g1[7] = 0;                                   // dim1_stride unused (2-D)
    v4i z4 = {0, 0, 0, 0};
    v8i z8 = {0, 0, 0, 0, 0, 0, 0, 0};
    __builtin_amdgcn_tensor_load_to_lds(g0, g1, z4, z4, z8, 0);
}

__global__ void fill_f32(float* p, float v, int n) {
    for (int i = blockIdx.x * blockDim.x + threadIdx.x; i < n;
         i += gridDim.x * blockDim.x)
        p[i] = v;
}

// ---------------- fp32 WMMA GEMM with TDM double-buffered LDS staging -------
// C[M,Nf] = A[M,K] @ B[K,Nf].  grid.x = M/16, block = 256 (8 waves).
// Wave w computes the 16 x 64 strip at columns [64w, 64w+64) (idle if >= Nf).
// Per K-chunk (KC=32): TDM loads A(16xKC) and B(KCxNf) tiles into LDS,
// double-buffered; wave 0 issues DMAs and gates on s_wait_tensorcnt.
// V_WMMA_F32_16X16X4_F32 fragments (32-bit layouts, CDNA5 ISA):
//   A 16x4: lane&15 = M, VGPR v holds K = v + 2*(lane>>4)
//   B 4x16: lane&15 = N, VGPR v holds K = v + 2*(lane>>4)
//   C 16x16: VGPR v -> row v + 8*(lane>>4), col = lane&15
__global__ __launch_bounds__(256) void gemm_wmma_async(
        const float* __restrict__ A, const float* __restrict__ B,
        float* __restrict__ C, int K, int Nf) {
    __shared__ float Ab[2][16 * KC];
    __shared__ float Bb[2][KC * NFMAX];

    const int tid  = threadIdx.x;
    const int wave = tid >> 5;
    const int lane = tid & 31;
    const int lm   = lane & 15;
    const int hi   = lane >> 4;
    const int m0   = blockIdx.x * 16;
    const int n0   = wave * 64;
    const bool active  = n0 < Nf;
    const int nchunks  = K / KC;

    if (wave == 0) {   // issue chunk 0
        tdm_load_2d(A + (size_t)m0 * K, lds_off(&Ab[0][0]), 16, KC,
                    (unsigned long long)K);
        tdm_load_2d(B, lds_off(&Bb[0][0]), KC, (unsigned)Nf,
                    (unsigned long long)Nf);
    }

    v8f acc[4] = {};
    for (int c = 0; c < nchunks; ++c) {
        const int cur = c & 1;
        if (wave == 0) {
            if (c + 1 < nchunks) {          // prefetch next chunk
                const int nxt = cur ^ 1;
                const int k0  = (c + 1) * KC;
                tdm_load_2d(A + (size_t)m0 * K + k0, lds_off(&Ab[nxt][0]),
                            16, KC, (unsigned long long)K);
                tdm_load_2d(B + (size_t)k0 * Nf, lds_off(&Bb[nxt][0]),
                            KC, (unsigned)Nf, (unsigned long long)Nf);
                __builtin_amdgcn_s_wait_tensorcnt(2);  // chunk c done (in-order)
            } else {
                __builtin_amdgcn_s_wait_tensorcnt(0);
            }
        }
        __syncthreads();
        if (active) {
            const float* Ac = Ab[cur];
            const float* Bc = Bb[cur];
#pragma unroll
            for (int kk = 0; kk < KC; kk += 4) {
                v2f a;
                a.x = Ac[lm * KC + kk + 2 * hi];
                a.y = Ac[lm * KC + kk + 2 * hi + 1];
#pragma unroll
                for (int j = 0; j < 4; ++j) {
                    const int col = n0 + j * 16 + lm;
                    v2f b;
                    b.x = Bc[(kk + 2 * hi) * Nf + col];
                    b.y = Bc[(kk + 1 + 2 * hi) * Nf + col];
                    acc[j] = __builtin_amdgcn_wmma_f32_16x16x4_f32(
                        false, a, false, b, (short)0, acc[j], false, false);
                }
            }
        }
        __syncthreads();
    }
    if (active) {
#pragma unroll
        for (int j = 0; j < 4; ++j)
#pragma unroll
            for (int v = 0; v < 8; ++v)
                C[(size_t)(m0 + v + 8 * hi) * Nf + n0 + j * 16 + lm] = acc[j][v];
    }
}

// ---------------- per-node attention logits ----------------
__global__ void node_logits(const float* __restrict__ h,
                            const float* __restrict__ a_src,
                            const float* __restrict__ a_dst,
                            float* __restrict__ als, float* __restrict__ ald,
                            int nN, int H, int C) {
    const int wave = blockIdx.x * (blockDim.x >> 5) + (threadIdx.x >> 5);
    const int lane = threadIdx.x & 31;
    if (wave >= nN) return;
    const int F = H * C;
    for (int hh = 0; hh < H; ++hh) {
        float s = 0.f, d = 0.f;
        for (int c = lane; c < C; c += 32) {
            float v = h[(size_t)wave * F + hh * C + c];
            s += v * a_src[hh * C + c];
            d += v * a_dst[hh * C + c];
        }
#pragma unroll
        for (int off = 16; off > 0; off >>= 1) {
            s += __shfl_down(s, off, 32);
            d += __shfl_down(d, off, 32);
        }
        if (lane == 0) {
            als[wave * H + hh] = s;
            ald[wave * H + hh] = d;
        }
    }
}

// ---------------- edge logits + LeakyReLU + segment max ----------------
__global__ void edge_max_k(const long long* __restrict__ ei,
                           const float* __restrict__ als,
                           const float* __restrict__ ald,
                           float* __restrict__ elog, float* __restrict__ maxb,
                           int nE, int nET, int H) {
    const int tid = blockIdx.x * blockDim.x + threadIdx.x;
    if (tid >= nET * H) return;
    const int e  = tid / H;
    const int hh = tid - e * H;
    int s, d;
    if (e < nE) { s = (int)ei[e]; d = (int)ei[nE + e]; }
    else        { s = d = e - nE; }
    float v = als[s * H + hh] + ald[d * H + hh];
    v = (v >= 0.f) ? v : 0.2f * v;          // LeakyReLU(0.2)
    elog[tid] = v;
    atomicMaxF(&maxb[d * H + hh], v);
}

// ---------------- exp(e - max) + segment sum ----------------
__global__ void edge_expsum_k(const long long* __restrict__ ei,
                              float* __restrict__ elog,
                              const float* __restrict__ maxb,
                              float* __restrict__ den,
                              int nE, int nET, int H) {
    const int tid = blockIdx.x * blockDim.x + threadIdx.x;
    if (tid >= nET * H) return;
    const int e  = tid / H;
    const int hh = tid - e * H;
    int d;
    if (e < nE) d = (int)ei[nE + e];
    else        d = e - nE;
    float ex = expf(elog[tid] - maxb[d * H + hh]);
    elog[tid] = ex;
    atomicAdd(&den[d * H + hh], ex);
}

// ---------------- message aggregation: out[dst] += h[src] * alpha ----------
__global__ void edge_aggregate_k(const long long* __restrict__ ei,
                                 const float* __restrict__ h,
                                 const float* __restrict__ elog,
                                 const float* __restrict__ den,
                                 float* __restrict__ out,
                                 int nE, int nET, int H, int C) {
    const int e = blockIdx.x;
    if (e >= nET) return;
    int s, d;
    if (e < nE) { s = (int)ei[e]; d = (int)ei[nE + e]; }
    else        { s = d = e - nE; }
    const int F = H * C;
    float alpha[2];
    for (int hh = 0; hh < H; ++hh)
        alpha[hh] = elog[e * H + hh] / den[d * H + hh];
    const float* hs = h + (size_t)s * F;
    float* od = out + (size_t)d * F;
    for (int c = threadIdx.x; c < F; c += blockDim.x) {
        const int hh = c / C;
        atomicAdd(&od[c], hs[c] * alpha[hh]);
    }
}

// ---------------- in-place bias + ReLU ----------------
__global__ void bias_relu_k(float* __restrict__ x, const float* __restrict__ b,
                            int total, int F) {
    for (int i = blockIdx.x * blockDim.x + threadIdx.x; i < total;
         i += gridDim.x * blockDim.x) {
        float v = x[i] + b[i % F];
        x[i] = v > 0.f ? v : 0.f;
    }
}

// ---------------- column-sum partials for mean pool ----------------
__global__ void col_mean_partial(const float* __restrict__ h,
                                 float* __restrict__ g, int nN, int F) {
    const int col = threadIdx.x;           // blockDim.x == F
    const int rows_per = (nN + gridDim.x - 1) / gridDim.x;
    const int r0 = blockIdx.x * rows_per;
    int r1 = r0 + rows_per;
    if (r1 > nN) r1 = nN;
    float s = 0.f;
    for (int r = r0; r < r1; ++r) s += h[(size_t)r * F + col];
    atomicAdd(&g[col], s);
}

// ---------------- final head: y = (g/N)@Wl + bl, then L2-normalize ----------
__global__ void final_head(const float* __restrict__ g,
                           const float* __restrict__ Wl,
                           const float* __restrict__ bl,
                           float* __restrict__ out, float invN) {
    __shared__ float sg[D2];
    __shared__ float ssum;
    const int j = threadIdx.x;             // blockDim.x == D2
    sg[j] = g[j] * invN;
    if (j == 0) ssum = 0.f;
    __syncthreads();
    float acc = bl[j];
    for (int i = 0; i < D2; ++i) acc += sg[i] * Wl[i * D2 + j];
    atomicAdd(&ssum, acc * acc);
    __syncthreads();
    float nrm = sqrtf(ssum);
    nrm = fmaxf(nrm, 1e-12f);
    out[j] = acc / nrm;
}

// ---------------- launch ----------------
extern "C" void kernel_launch(void* const* d_in, const int* in_sizes, int n_in,
                              void* d_out, int out_size, void* d_ws, size_t ws_size,
                              hipStream_t stream) {
    (void)in_sizes; (void)n_in; (void)out_size; (void)ws_size;
    const float*     x      = (const float*)d_in[0];
    const long long* ei     = (const long long*)d_in[1];   // int64 [2,E]
    const float*     W1     = (const float*)d_in[2];
    const float*     a_src1 = (const float*)d_in[3];
    const float*     a_dst1 = (const float*)d_in[4];
    const float*     b1     = (const float*)d_in[5];
    const float*     W2     = (const float*)d_in[6];
    const float*     a_src2 = (const float*)d_in[7];
    const float*     a_dst2 = (const float*)d_in[8];
    const float*     b2     = (const float*)d_in[9];
    const float*     Wl     = (const float*)d_in[10];
    const float*     bl     = (const float*)d_in[11];
    float* outp = (float*)d_out;

    // ---- carve workspace (floats) ----
    float* p = (float*)d_ws;
    float* h1    = p;  p += (size_t)NN * F1;   // GEMM1 out
    float* out1  = p;  p += (size_t)NN * F1;   // layer1 agg (ReLU'd in place)
    float* h2g   = p;  p += (size_t)NN * D2;   // GEMM2 out
    float* out2  = p;  p += (size_t)NN * D2;   // layer2 agg (ReLU'd in place)
    float* elog1 = p;  p += (size_t)ET * H1;
    float* elog2 = p;  p += (size_t)ET;
    float* als1  = p;  p += (size_t)NN * H1;
    float* ald1  = p;  p += (size_t)NN * H1;
    float* max1  = p;  p += (size_t)NN * H1;
    float* den1  = p;  p += (size_t)NN * H1;
    float* als2  = p;  p += (size_t)NN;
    float* ald2  = p;  p += (size_t)NN;
    float* max2  = p;  p += (size_t)NN;
    float* den2  = p;  p += (size_t)NN;
    float* g     = p;  p += D2;

    const float NEG_INF = -__builtin_huge_valf();

    // ---- init accumulators (every call) ----
    fill_f32<<<64, 256, 0, stream>>>(max1, NEG_INF, NN * H1);
    fill_f32<<<64, 256, 0, stream>>>(den1, 0.f, NN * H1);
    fill_f32<<<2048, 256, 0, stream>>>(out1, 0.f, NN * F1);
    fill_f32<<<64, 256, 0, stream>>>(max2, NEG_INF, NN);
    fill_f32<<<64, 256, 0, stream>>>(den2, 0.f, NN);
    fill_f32<<<2048, 256, 0, stream>>>(out2, 0.f, NN * D2);
    fill_f32<<<1, 256, 0, stream>>>(g, 0.f, D2);

    // ================= layer 1 (heads=2, C=256) =================
    gemm_wmma_async<<<NN / 16, 256, 0, stream>>>(x, W1, h1, DIN, F1);
    node_logits<<<(NN + 7) / 8, 256, 0, stream>>>(h1, a_src1, a_dst1,
                                                  als1, ald1, NN, H1, C1);
    edge_max_k<<<(ET * H1 + 255) / 256, 256, 0, stream>>>(ei, als1, ald1,
                                                          elog1, max1, EE, ET, H1);
    edge_expsum_k<<<(ET * H1 + 255) / 256, 256, 0, stream>>>(ei, elog1, max1,
                                                             den1, EE, ET, H1);
    edge_aggregate_k<<<ET, 128, 0, stream>>>(ei, h1, elog1, den1, out1,
                                             EE, ET, H1, C1);
    bias_relu_k<<<2048, 256, 0, stream>>>(out1, b1, NN * F1, F1);

    // ================= layer 2 (heads=1, C=384) =================
    gemm_wmma_async<<<NN / 16, 256, 0, stream>>>(out1, W2, h2g, F1, D2);
    node_logits<<<(NN + 7) / 8, 256, 0, stream>>>(h2g, a_src2, a_dst2,
                                                  als2, ald2, NN, 1, D2);
    edge_max_k<<<(ET + 255) / 256, 256, 0, stream>>>(ei, als2, ald2,
                                                     elog2, max2, EE, ET, 1);
    edge_expsum_k<<<(ET + 255) / 256, 256, 0, stream>>>(ei, elog2, max2,
                                                        den2, EE, ET, 1);
    edge_aggregate_k<<<ET, 128, 0, stream>>>(ei, h2g, elog2, den2, out2,
                                             EE, ET, 1, D2);
    bias_relu_k<<<2048, 256, 0, stream>>>(out2, b2, NN * D2, D2);

    // ================= pool + head + normalize =================
    col_mean_partial<<<80, D2, 0, stream>>>(out2, g, NN, D2);
    final_head<<<1, D2, 0, stream>>>(g, Wl, bl, outp, 1.0f / (float)NN);
}